// RNforPixel_38817914421368
// MI455X (gfx1250) — compile-verified
//
#include <hip/hip_runtime.h>
#include <hip/hip_bf16.h>

typedef float v2f __attribute__((ext_vector_type(2)));
typedef float v8f __attribute__((ext_vector_type(8)));

// ---------------------------------------------------------------------------
// f32 WMMA 16x16x4. A: 16x4 (2 VGPRs/lane), B: 4x16, C/D: 16x16 (8 VGPRs).
// Lanes 0-15 hold K=k,k+1; lanes 16-31 hold K=k+2,k+3 (ISA 7.12.2).
// D: VGPR i -> row (i + 8*(lane>=16)), col = lane&15.
// ---------------------------------------------------------------------------
__device__ inline v8f wmma4(v2f a, v2f b, v8f c) {
    return __builtin_amdgcn_wmma_f32_16x16x4_f32(
        false, a, false, b, (short)0, c, false, false);
}

__device__ inline float sigm(float x) { return 1.0f / (1.0f + expf(-x)); }
__device__ inline int imin(int a, int b) { return a < b ? a : b; }

// ---------------------------------------------------------------------------
// Generic f32 WMMA GEMM: C[M,N] = A[M,K] @ B[K,N] (+ bias row) (optional relu)
// 256 threads = 8 waves, one 16x16 tile per wave (flattened tile index).
// Requires: K % 4 == 0, lda even. Out-of-range rows/cols are CLAMPED for
// loads (garbage only pollutes unstored D rows/cols). All hot-loop address
// math is 32-bit offsets with constant increments.
// ---------------------------------------------------------------------------
__global__ void gemm_f32_wmma(const float* __restrict__ A, const float* __restrict__ B,
                              const float* __restrict__ bias, float* __restrict__ C,
                              int M, int N, int K, int lda, int ldb, int ldc, int relu)
{
    const int lane = threadIdx.x & 31;
    const int wave = threadIdx.x >> 5;
    const int half = lane >> 4;
    const int l16  = lane & 15;

    const int mtiles = (M + 15) >> 4;
    const int ntiles = (N + 15) >> 4;
    const int tile = blockIdx.x * 8 + wave;
    if (tile >= mtiles * ntiles) return;
    const int mt = tile / ntiles;
    const int nt = tile - mt * ntiles;
    const int m0 = mt * 16, n0 = nt * 16;

    const int row  = m0 + l16;
    const int col  = n0 + l16;
    const int rowL = imin(row, M - 1);   // clamped for loads
    const int colL = imin(col, N - 1);

    const float* Arow = A + rowL * lda + half * 2;   // lane's A base
    const float* Bcol = B + colL;                    // lane's B column base
    int bOff = half * 2 * ldb;                       // 32-bit running offset
    const int bInc = 4 * ldb;

    v8f acc = {};
    #pragma unroll 4
    for (int k = 0; k < K; k += 4) {
        const v2f af = *(const v2f*)(Arow + k);      // global_load_b64
        v2f bf;
        bf.x = Bcol[bOff];
        bf.y = Bcol[bOff + ldb];
        bOff += bInc;
        acc = wmma4(af, bf, acc);
    }
    if (col < N) {
        const float bv = bias ? bias[col] : 0.0f;
        #pragma unroll
        for (int i = 0; i < 8; ++i) {
            const int r = m0 + i + half * 8;
            if (r < M) {
                float v = acc[i] + bv;
                if (relu) v = fmaxf(v, 0.0f);
                C[r * ldc + col] = v;
            }
        }
    }
}

// ---------------------------------------------------------------------------
// Conv 3x3 stride2 pad1 + bias + ReLU as on-the-fly im2col WMMA GEMM.
// M=4096 pixels (b*64 + orow*8 + ocol), N=24 ch (pad 32), K=4608 (ci*9+kh*3+kw).
// Zero padding via clamp-address + select (unconditional loads). Weight pair
// is one b64 load. k->(ci,r9) decode is incremental/branchless, 32-bit offsets.
// 256 threads = 8 waves, one tile per wave; grid.x = 512/8 = 64 blocks.
// ---------------------------------------------------------------------------
__global__ void conv_relu_wmma(const float* __restrict__ img, const float* __restrict__ conv_w,
                               const float* __restrict__ conv_b, float* __restrict__ y)
{
    const int lane = threadIdx.x & 31;
    const int wave = threadIdx.x >> 5;
    const int half = lane >> 4;
    const int l16  = lane & 15;

    const int tile = blockIdx.x * 8 + wave;   // 0..511
    const int mt = tile >> 1;                 // 0..255
    const int nt = tile & 1;                  // 0..1
    const int m  = mt * 16 + l16;
    const int b    = m >> 6;
    const int p    = m & 63;
    const int orow = p >> 3;
    const int ocol = p & 7;
    const int cch  = nt * 16 + l16;
    const int cchc = imin(cch, 23);

    const float* imgb = img + b * (512 * 256);           // per-lane image base
    const float* wrow = conv_w + cchc * 4608 + half * 2; // lane's weight base

    // Per-lane spatial bases for the 3x3 window (clamped) + validity.
    // ir = orow*2-1+kh, ic = ocol*2-1+kw, kh/kw in {0,1,2}.
    int soff[3][3];      // clamped irc*16+icc
    int svalid[3][3];
    #pragma unroll
    for (int kh = 0; kh < 3; ++kh) {
        #pragma unroll
        for (int kw = 0; kw < 3; ++kw) {
            const int ir = orow * 2 - 1 + kh;
            const int ic = ocol * 2 - 1 + kw;
            const int irc = ir < 0 ? 0 : (ir > 15 ? 15 : ir);
            const int icc = ic < 0 ? 0 : (ic > 15 ? 15 : ic);
            soff[kh][kw]   = irc * 16 + icc;
            svalid[kh][kw] = ((unsigned)ir < 16u) & ((unsigned)ic < 16u);
        }
    }

    int ci = 0, r9 = half * 2;               // incremental k decode, r9 in [0,8]
    int cOff = 0;                            // ci*256, 32-bit running offset
    v8f acc = {};
    #pragma unroll 3
    for (int k = 0; k < 4608; k += 4) {
        const v2f bf = *(const v2f*)(wrow + k);          // global_load_b64
        v2f af;
        #pragma unroll
        for (int t = 0; t < 2; ++t) {
            int co = cOff, r = r9 + t;
            if (t == 1 && r == 9) { co = cOff + 256; r = 0; }
            const int kh = (r >= 6) ? 2 : ((r >= 3) ? 1 : 0);
            const int kw = r - kh * 3;
            float a = imgb[co + soff[kh][kw]];           // always in-bounds
            a = svalid[kh][kw] ? a : 0.0f;
            if (t == 0) af.x = a; else af.y = a;
        }
        acc = wmma4(af, bf, acc);
        r9 += 4;
        const int wrap = (r9 >= 9);
        ci += wrap;
        cOff += wrap ? 256 : 0;
        r9 -= wrap ? 9 : 0;
    }
    if (cch < 24) {
        const float cb = conv_b[cch];
        #pragma unroll
        for (int i = 0; i < 8; ++i) {
            const int rr = mt * 16 + i + half * 8;
            const int bb = rr >> 6, pp = rr & 63;
            y[(bb * 24 + cch) * 64 + pp] = fmaxf(acc[i] + cb, 0.0f);
        }
    }
}

// ---------------------------------------------------------------------------
// BN stats (mean/var over B*H*W per channel) fused with per-batch sums:
// s[b][c] = bn_g*rstd*(sum_p y[b,c,p] - 64*mean) + 64*bn_b
// s[b][24], s[b][25] = coord-channel sums = 0.
// ---------------------------------------------------------------------------
__global__ void bn_s_kernel(const float* __restrict__ y, const float* __restrict__ bn_g,
                            const float* __restrict__ bn_b, float* __restrict__ s)
{
    const int c = blockIdx.x;   // 0..23
    const int b = threadIdx.x;  // 0..63
    float sum = 0.0f, sum2 = 0.0f;
    for (int p = 0; p < 64; ++p) {
        const float v = y[(b * 24 + c) * 64 + p];
        sum += v; sum2 += v * v;
    }
    __shared__ float ls[64], ls2[64];
    ls[b] = sum; ls2[b] = sum2;
    __syncthreads();
    for (int off = 32; off > 0; off >>= 1) {
        if (b < off) { ls[b] += ls[b + off]; ls2[b] += ls2[b + off]; }
        __syncthreads();
    }
    const float mean = ls[0] * (1.0f / 4096.0f);
    const float var  = ls2[0] * (1.0f / 4096.0f) - mean * mean;
    const float rstd = rsqrtf(var + 1e-5f);
    s[b * 26 + c] = bn_g[c] * rstd * (sum - 64.0f * mean) + 64.0f * bn_b[c];
    if (c == 0) { s[b * 26 + 24] = 0.0f; s[b * 26 + 25] = 0.0f; }
}

// Pack combined LSTM weight: Wz[k][n], k<32 -> w_ih[n][k], else w_hh[n][k-32].
__global__ void wz_kernel(const float* __restrict__ w_ih, const float* __restrict__ w_hh,
                          float* __restrict__ Wz)
{
    const int idx = blockIdx.x * blockDim.x + threadIdx.x;
    if (idx >= 160 * 512) return;
    const int k = idx / 512, n = idx - k * 512;
    Wz[idx] = (k < 32) ? w_ih[n * 32 + k] : w_hh[n * 128 + (k - 32)];
}

// ---------------------------------------------------------------------------
// LSTM: 4 blocks x 16 batches, 256 threads (8 waves). Per step:
// z[16,512] = [x_t | h][16,160] @ Wz[160,512] + (b_ih+b_hh) via WMMA from LDS,
// gates (i,f,g,o) in VALU updating LDS-resident h,c. 20 steps.
// ---------------------------------------------------------------------------
__global__ void lstm_kernel(const int* __restrict__ qu, const float* __restrict__ emb_w,
                            const float* __restrict__ Wz, const float* __restrict__ b_ih,
                            const float* __restrict__ b_hh, float* __restrict__ qst)
{
    __shared__ float xh[16 * 160];    // [x_t(32) | h(128)] per batch row
    __shared__ float zbuf[16 * 512];
    __shared__ float cst[16 * 128];
    const int tid = threadIdx.x;
    const int b0 = blockIdx.x * 16;
    for (int i = tid; i < 16 * 128; i += 256) cst[i] = 0.0f;
    for (int i = tid; i < 16 * 160; i += 256) xh[i] = 0.0f;
    __syncthreads();

    const int lane = tid & 31, wave = tid >> 5;
    const int half = lane >> 4, l16 = lane & 15;
    const float* xrow = xh + l16 * 160 + half * 2;

    for (int t = 0; t < 20; ++t) {
        for (int i = tid; i < 16 * 32; i += 256) {
            const int m = i >> 5, e = i & 31;
            const int q = qu[(b0 + m) * 20 + t];
            xh[m * 160 + e] = emb_w[q * 32 + e];
        }
        __syncthreads();
        #pragma unroll
        for (int tile = 0; tile < 4; ++tile) {
            const int n0 = wave * 64 + tile * 16;
            const float* Wc = Wz + n0 + l16 + half * 2 * 512;
            int wOff = 0;
            v8f acc = {};
            #pragma unroll 4
            for (int k = 0; k < 160; k += 4) {
                const v2f af = *(const v2f*)(xrow + k);   // ds_load_b64
                v2f bf;
                bf.x = Wc[wOff];
                bf.y = Wc[wOff + 512];
                wOff += 4 * 512;
                acc = wmma4(af, bf, acc);
            }
            const int col = n0 + l16;
            const float bv = b_ih[col] + b_hh[col];
            #pragma unroll
            for (int i = 0; i < 8; ++i)
                zbuf[(i + half * 8) * 512 + col] = acc[i] + bv;
        }
        __syncthreads();
        for (int i = tid; i < 16 * 128; i += 256) {
            const int m = i >> 7, j = i & 127;
            const float zi = zbuf[m * 512 + j];
            const float zf = zbuf[m * 512 + 128 + j];
            const float zg = zbuf[m * 512 + 256 + j];
            const float zo = zbuf[m * 512 + 384 + j];
            const float cc = sigm(zf) * cst[i] + sigm(zi) * tanhf(zg);
            cst[i] = cc;
            xh[m * 160 + 32 + j] = sigm(zo) * tanhf(cc);
        }
        __syncthreads();
    }
    for (int i = tid; i < 16 * 128; i += 256) {
        const int m = i >> 7, j = i & 127;
        qst[(b0 + m) * 128 + j] = xh[m * 160 + 32 + j];
    }
}

// cvec[j] = 4096 * (b2@W34 + b3@g_w4 + b4)[j]
__global__ void cvec_kernel(const float* __restrict__ b2, const float* __restrict__ W34,
                            const float* __restrict__ b3, const float* __restrict__ g_w4,
                            const float* __restrict__ b4, float* __restrict__ cvec)
{
    const int j = threadIdx.x;
    float a = b4[j];
    for (int k = 0; k < 256; ++k)
        a += b2[k] * W34[k * 256 + j] + b3[k] * g_w4[k * 256 + j];
    cvec[j] = 4096.0f * a;
}

// u[b] = 64 * s[b]@(W1a + W1b) + 4096 * (qst[b]@W1q + b1)   == Sum_{i,j} h[b,i,j]
__global__ void u_kernel(const float* __restrict__ s, const float* __restrict__ qst,
                         const float* __restrict__ g_w1, const float* __restrict__ g_b1,
                         float* __restrict__ u)
{
    const int b = blockIdx.x, j = threadIdx.x;
    float acc = 0.0f;
    for (int c = 0; c < 26; ++c)
        acc += s[b * 26 + c] * (g_w1[c * 256 + j] + g_w1[(26 + c) * 256 + j]);
    acc *= 64.0f;
    float aq = g_b1[j];
    for (int h = 0; h < 128; ++h)
        aq += qst[b * 128 + h] * g_w1[(52 + h) * 256 + j];
    u[b * 256 + j] = acc + 4096.0f * aq;
}

extern "C" void kernel_launch(void* const* d_in, const int* in_sizes, int n_in,
                              void* d_out, int out_size, void* d_ws, size_t ws_size,
                              hipStream_t stream) {
    const float* img    = (const float*)d_in[0];
    const int*   qu     = (const int*)  d_in[1];
    const float* conv_w = (const float*)d_in[2];
    const float* conv_b = (const float*)d_in[3];
    const float* bn_g   = (const float*)d_in[4];
    const float* bn_b   = (const float*)d_in[5];
    const float* emb_w  = (const float*)d_in[6];
    const float* w_ih   = (const float*)d_in[7];
    const float* w_hh   = (const float*)d_in[8];
    const float* b_ih   = (const float*)d_in[9];
    const float* b_hh   = (const float*)d_in[10];
    const float* g_w1   = (const float*)d_in[11];
    const float* g_b1   = (const float*)d_in[12];
    const float* g_w2   = (const float*)d_in[13];
    const float* g_b2   = (const float*)d_in[14];
    const float* g_w3   = (const float*)d_in[15];
    const float* g_b3   = (const float*)d_in[16];
    const float* g_w4   = (const float*)d_in[17];
    const float* g_b4   = (const float*)d_in[18];
    const float* f_w1   = (const float*)d_in[19];
    const float* f_b1   = (const float*)d_in[20];
    const float* f_w2   = (const float*)d_in[21];
    const float* f_b2   = (const float*)d_in[22];
    const float* f_w3   = (const float*)d_in[23];
    const float* f_b3   = (const float*)d_in[24];
    float* out = (float*)d_out;

    float* ws = (float*)d_ws;
    float* ws_y    = ws;                  // 64*24*64 = 98304
    float* ws_s    = ws_y    + 98304;     // 64*26    = 1664
    float* ws_Wz   = ws_s    + 1664;      // 160*512  = 81920
    float* ws_qst  = ws_Wz   + 81920;     // 64*128   = 8192
    float* ws_W34  = ws_qst  + 8192;      // 256*256  = 65536
    float* ws_W234 = ws_W34  + 65536;     // 256*256  = 65536
    float* ws_cvec = ws_W234 + 65536;     // 256
    float* ws_u    = ws_cvec + 256;       // 64*256   = 16384
    float* ws_xg   = ws_u    + 16384;     // 64*256
    float* ws_y1   = ws_xg   + 16384;     // 64*256
    float* ws_y2   = ws_y1   + 16384;     // 64*256

    // 1) conv + relu (WMMA im2col GEMM): 512 tiles, 8 waves/block
    conv_relu_wmma<<<64, 256, 0, stream>>>(img, conv_w, conv_b, ws_y);

    // 2) BN stats + collapsed per-batch channel sums -> s[64,26]
    bn_s_kernel<<<24, 64, 0, stream>>>(ws_y, bn_g, bn_b, ws_s);

    // 3) LSTM path
    wz_kernel<<<(160 * 512 + 255) / 256, 256, 0, stream>>>(w_ih, w_hh, ws_Wz);
    lstm_kernel<<<4, 256, 0, stream>>>(qu, emb_w, ws_Wz, b_ih, b_hh, ws_qst);

    // 4) fold linear g-layers: W34 = W3@W4, W234 = W2@W34, cvec (256 tiles -> 32 blocks)
    gemm_f32_wmma<<<32, 256, 0, stream>>>(g_w3, g_w4, nullptr, ws_W34,
                                          256, 256, 256, 256, 256, 256, 0);
    gemm_f32_wmma<<<32, 256, 0, stream>>>(g_w2, ws_W34, nullptr, ws_W234,
                                          256, 256, 256, 256, 256, 256, 0);
    cvec_kernel<<<1, 256, 0, stream>>>(g_b2, ws_W34, g_b3, g_w4, g_b4, ws_cvec);

    // 5) u = Sum_{i,j} h[b,i,j] (collapsed first g-layer), x_g = u@W234 + cvec
    u_kernel<<<64, 256, 0, stream>>>(ws_s, ws_qst, g_w1, g_b1, ws_u);
    gemm_f32_wmma<<<8, 256, 0, stream>>>(ws_u, ws_W234, ws_cvec, ws_xg,
                                         64, 256, 256, 256, 256, 256, 0);

    // 6) fcout: relu, relu, final (64x28)
    gemm_f32_wmma<<<8, 256, 0, stream>>>(ws_xg, f_w1, f_b1, ws_y1,
                                         64, 256, 256, 256, 256, 256, 1);
    gemm_f32_wmma<<<8, 256, 0, stream>>>(ws_y1, f_w2, f_b2, ws_y2,
                                         64, 256, 256, 256, 256, 256, 1);
    gemm_f32_wmma<<<1, 256, 0, stream>>>(ws_y2, f_w3, f_b3, out,
                                         64, 28, 256, 256, 28, 28, 0);
    (void)in_sizes; (void)n_in; (void)out_size; (void)ws_size;
}